// GIN_4561255269295
// MI455X (gfx1250) — compile-verified
//
#include <hip/hip_runtime.h>
#include <hip/hip_bf16.h>
#include <math.h>

// ---------------- problem constants (match reference) ----------------
#define NN   50000
#define EE   1600000
#define FIN  3
#define HH   64
#define GG   512
#define CC   2
#define BN_EPS 1e-5f

typedef __attribute__((ext_vector_type(2))) float v2f;
typedef __attribute__((ext_vector_type(8))) float v8f;

// ---------------- simple helpers ----------------
__global__ void zero_f32(float* p, int n) {
    int i = blockIdx.x * blockDim.x + threadIdx.x;
    if (i < n) p[i] = 0.0f;
}

__global__ void copy_f32(const float* __restrict__ s, float* __restrict__ d, int n) {
    int i = blockIdx.x * blockDim.x + threadIdx.x;
    if (i < n) d[i] = s[i];
}

// Fold BatchNorm (eval) + preceding bias into per-channel scale/shift:
// y = ((acc + b1) - m) * g*rsqrt(v+eps) + bb  ==  acc*scale + shift
__global__ void bn_prep(const float* __restrict__ b1, const float* __restrict__ g,
                        const float* __restrict__ bb, const float* __restrict__ m,
                        const float* __restrict__ v, float* __restrict__ scale,
                        float* __restrict__ shift) {
    int f = threadIdx.x;           // launched with HH threads
    float s = g[f] * rsqrtf(v[f] + BN_EPS);
    scale[f] = s;
    shift[f] = (b1[f] - m[f]) * s + bb[f];
}

// ---------------- edge scatter-add (h0 must be pre-initialized to x) ----------------
__global__ void scatter_add3(const float* __restrict__ x, const int* __restrict__ src,
                             const int* __restrict__ dst, float* __restrict__ h0) {
    int e = blockIdx.x * blockDim.x + threadIdx.x;
    if (e >= EE) return;
    int s = src[e] * FIN, d = dst[e] * FIN;
    atomicAdd(h0 + d + 0, x[s + 0]);
    atomicAdd(h0 + d + 1, x[s + 1]);
    atomicAdd(h0 + d + 2, x[s + 2]);
}

__global__ void scatter_add64(const float* __restrict__ x, const int* __restrict__ src,
                              const int* __restrict__ dst, float* __restrict__ h0) {
    int t = blockIdx.x * blockDim.x + threadIdx.x;
    int e = t >> 2;
    if (e >= EE) return;
    int c = (t & 3) * 16;                       // 16 contiguous features per thread
    int s = src[e], d = dst[e];
    const float4* xs = (const float4*)(x + s * HH + c);
    float* hd = h0 + d * HH + c;
#pragma unroll
    for (int i = 0; i < 4; ++i) {
        float4 v = xs[i];
        atomicAdd(hd + i * 4 + 0, v.x);
        atomicAdd(hd + i * 4 + 1, v.y);
        atomicAdd(hd + i * 4 + 2, v.z);
        atomicAdd(hd + i * 4 + 3, v.w);
    }
}

// ---------------- layer-1 first linear: [N x 3] @ [3 x 64], BN+ReLU fused ----------------
__global__ void lin3_bn_relu(const float* __restrict__ h0, const float* __restrict__ w1,
                             const float* __restrict__ scale, const float* __restrict__ shift,
                             float* __restrict__ out) {
    int t = blockIdx.x * blockDim.x + threadIdx.x;
    if (t >= NN * HH) return;
    int n = t >> 6, f = t & 63;
    const float* h = h0 + n * FIN;
    float acc = h[0] * w1[f] + h[1] * w1[HH + f] + h[2] * w1[2 * HH + f];
    out[t] = fmaxf(acc * scale[f] + shift[f], 0.0f);
}

// ---------------- fp32 WMMA GEMM: Out[M x Nc] = epi(A[M x K] @ W[K x Nc]) ----------------
// Block = 128 threads = 4 waves; block computes a 16-row x 64-col strip.
// Stage A strip (16 x K) and W column slice (K x 64) in LDS once, then the
// steady-state loop is 2 x ds_load_b64 + v_wmma_f32_16x16x4_f32 per K-step.
// LDS rows padded to K+4 floats: float4 staging stays 16B aligned, fragment
// reads hit banks (4*row + k) % 64 -> conflict-free across a half-wave, and
// ds_load_b64 stays 8B aligned (k-phase is even).
// A frag (16x4 f32): lanes 0-15 hold K=k0,k0+1; lanes 16-31 hold K=k0+2,k0+3.
// B frag mirrored. C/D: VGPR r -> row mBase+r+(lane>>4)*8, col nBase+(lane&15).
// Requires: M % 16 == 0, Nc % 64 == 0, K % 4 == 0 (all hold here).
template <int KK, bool RELU>
__global__ void __launch_bounds__(128)
gemm_wmma_fused(const float* __restrict__ A, const float* __restrict__ W,
                const float* __restrict__ scale, const float* __restrict__ shift,
                float* __restrict__ Out, int Nc) {
    constexpr int KP = KK + 4;                  // padded LDS row stride (floats)
    __shared__ float Alds[16 * KP];
    __shared__ float Wlds[64 * KP];

    const int tid   = threadIdx.x;
    const int lane  = tid & 31;
    const int wave  = tid >> 5;
    const int mBase = blockIdx.x * 16;
    const int cBase = blockIdx.y * 64;

    // ---- stage A strip [16 x KK] via float4 (global coalesced, LDS aligned) ----
    constexpr int AQ = 16 * (KK / 4);
    for (int idx = tid; idx < AQ; idx += 128) {
        int r  = idx / (KK / 4);
        int k4 = (idx % (KK / 4)) * 4;
        float4 v = *(const float4*)(A + (mBase + r) * KK + k4);
        *(float4*)(&Alds[r * KP + k4]) = v;
    }
    // ---- stage W slice [KK x 64] transposed into Wlds[c * KP + k] ----
    for (int idx = tid; idx < KK * 64; idx += 128) {
        int k = idx >> 6;
        int c = idx & 63;                       // consecutive tid -> coalesced global
        Wlds[c * KP + k] = W[k * Nc + cBase + c];
    }
    __syncthreads();

    const int lc  = lane & 15;
    const int kPh = (lane >> 4) * 2;            // 0 for lanes 0-15, 2 for lanes 16-31
    const float* aBase = &Alds[lc * KP + kPh];
    const float* bBase = &Wlds[(wave * 16 + lc) * KP + kPh];

    v8f acc = {};
#pragma unroll
    for (int k0 = 0; k0 < KK; k0 += 4) {
        v2f a = *(const v2f*)(aBase + k0);      // ds_load_b64
        v2f b = *(const v2f*)(bBase + k0);      // ds_load_b64
        acc = __builtin_amdgcn_wmma_f32_16x16x4_f32(
            /*neg_a=*/false, a, /*neg_b=*/false, b,
            /*c_mod=*/(short)0, acc, /*reuse_a=*/false, /*reuse_b=*/false);
    }

    const int col = cBase + wave * 16 + lc;
    const float s = scale ? scale[col] : 1.0f;
    const float t = shift[col];
#pragma unroll
    for (int r = 0; r < 8; ++r) {
        int orow = mBase + r + (lane >> 4) * 8;
        float v = acc[r] * s + t;
        if (RELU) v = fmaxf(v, 0.0f);
        Out[orow * Nc + col] = v;
    }
}

// ---------------- global_add_pool: pooled[batch[n], colOff + f] += h[n, f] ----------------
__global__ void pool_add(const float* __restrict__ h, const int* __restrict__ batch,
                         float* __restrict__ pooled, int colOff) {
    int t = blockIdx.x * blockDim.x + threadIdx.x;
    int n = t >> 1;
    if (n >= NN) return;
    int c = (t & 1) * 32;
    int g = batch[n];
    const float* hn = h + n * HH + c;
    float* pg = pooled + g * (3 * HH) + colOff + c;
#pragma unroll
    for (int i = 0; i < 32; ++i) atomicAdd(pg + i, hn[i]);
}

// ---------------- head: lin2 [192 x 2] + log_softmax; writes (h, log_softmax) ----------------
__global__ void head_kernel(const float* __restrict__ q, const float* __restrict__ w2,
                            const float* __restrict__ b2, float* __restrict__ out) {
    int g = blockIdx.x * blockDim.x + threadIdx.x;
    if (g >= GG) return;
    const float* qg = q + g * (3 * HH);
    float c0 = b2[0], c1 = b2[1];
#pragma unroll 8
    for (int k = 0; k < 3 * HH; ++k) {
        float qv = qg[k];
        c0 += qv * w2[k * CC + 0];
        c1 += qv * w2[k * CC + 1];
    }
    out[g * CC + 0] = c0;
    out[g * CC + 1] = c1;
    float mx  = fmaxf(c0, c1);
    float lse = mx + logf(expf(c0 - mx) + expf(c1 - mx));
    out[GG * CC + g * CC + 0] = c0 - lse;
    out[GG * CC + g * CC + 1] = c1 - lse;
}

// =============================== launcher ===============================
extern "C" void kernel_launch(void* const* d_in, const int* in_sizes, int n_in,
                              void* d_out, int out_size, void* d_ws, size_t ws_size,
                              hipStream_t stream) {
    const float* x      = (const float*)d_in[0];
    // conv layer l (1..3): base = 1 + (l-1)*8 : w1 b1 bn_g bn_b bn_m bn_v w2 b2
    const float* lin1_w = (const float*)d_in[25];
    const float* lin1_b = (const float*)d_in[26];
    const float* lin2_w = (const float*)d_in[27];
    const float* lin2_b = (const float*)d_in[28];
    const int*   edge   = (const int*)d_in[29];
    const int*   src    = edge;
    const int*   dst    = edge + EE;
    const int*   batch  = (const int*)d_in[30];

    // workspace layout (floats)
    float* p     = (float*)d_ws;
    float* h0_1  = p; p += (size_t)NN * FIN;   // x + agg (layer 1)
    float* h0    = p; p += (size_t)NN * HH;    // h + agg (layers 2,3)
    float* tmp   = p; p += (size_t)NN * HH;    // after first MLP linear
    float* h1    = p; p += (size_t)NN * HH;
    float* h2    = p; p += (size_t)NN * HH;
    float* h3    = p; p += (size_t)NN * HH;
    float* pooled= p; p += (size_t)GG * 3 * HH;
    float* q     = p; p += (size_t)GG * 3 * HH;
    float* bnS   = p; p += HH;
    float* bnT   = p; p += HH;

    const int TB = 256;
    dim3 blk(TB);

    // zero the pooled accumulator every call (ws is not re-poisoned/zeroed)
    zero_f32<<<dim3((GG * 3 * HH + TB - 1) / TB), blk, 0, stream>>>(pooled, GG * 3 * HH);

    dim3 gemmBlk(128);
    dim3 gemmGridH(NN / 16, 1);                // Nc = 64 -> one 64-col strip
    dim3 scat64Grid((EE * 4 + TB - 1) / TB);
    dim3 poolGrid((NN * 2 + TB - 1) / TB);

    // ---------------- conv1 ----------------
    copy_f32<<<dim3((NN * FIN + TB - 1) / TB), blk, 0, stream>>>(x, h0_1, NN * FIN);
    scatter_add3<<<dim3((EE + TB - 1) / TB), blk, 0, stream>>>(x, src, dst, h0_1);
    bn_prep<<<1, HH, 0, stream>>>((const float*)d_in[2], (const float*)d_in[3],
                                  (const float*)d_in[4], (const float*)d_in[5],
                                  (const float*)d_in[6], bnS, bnT);
    lin3_bn_relu<<<dim3((NN * HH + TB - 1) / TB), blk, 0, stream>>>(
        h0_1, (const float*)d_in[1], bnS, bnT, tmp);
    gemm_wmma_fused<HH, true><<<gemmGridH, gemmBlk, 0, stream>>>(
        tmp, (const float*)d_in[7], nullptr, (const float*)d_in[8], h1, HH);
    pool_add<<<poolGrid, blk, 0, stream>>>(h1, batch, pooled, 0);

    // ---------------- conv2 ----------------
    copy_f32<<<dim3((NN * HH + TB - 1) / TB), blk, 0, stream>>>(h1, h0, NN * HH);
    scatter_add64<<<scat64Grid, blk, 0, stream>>>(h1, src, dst, h0);
    bn_prep<<<1, HH, 0, stream>>>((const float*)d_in[10], (const float*)d_in[11],
                                  (const float*)d_in[12], (const float*)d_in[13],
                                  (const float*)d_in[14], bnS, bnT);
    gemm_wmma_fused<HH, true><<<gemmGridH, gemmBlk, 0, stream>>>(
        h0, (const float*)d_in[9], bnS, bnT, tmp, HH);
    gemm_wmma_fused<HH, true><<<gemmGridH, gemmBlk, 0, stream>>>(
        tmp, (const float*)d_in[15], nullptr, (const float*)d_in[16], h2, HH);
    pool_add<<<poolGrid, blk, 0, stream>>>(h2, batch, pooled, HH);

    // ---------------- conv3 ----------------
    copy_f32<<<dim3((NN * HH + TB - 1) / TB), blk, 0, stream>>>(h2, h0, NN * HH);
    scatter_add64<<<scat64Grid, blk, 0, stream>>>(h2, src, dst, h0);
    bn_prep<<<1, HH, 0, stream>>>((const float*)d_in[18], (const float*)d_in[19],
                                  (const float*)d_in[20], (const float*)d_in[21],
                                  (const float*)d_in[22], bnS, bnT);
    gemm_wmma_fused<HH, true><<<gemmGridH, gemmBlk, 0, stream>>>(
        h0, (const float*)d_in[17], bnS, bnT, tmp, HH);
    gemm_wmma_fused<HH, true><<<gemmGridH, gemmBlk, 0, stream>>>(
        tmp, (const float*)d_in[23], nullptr, (const float*)d_in[24], h3, HH);
    pool_add<<<poolGrid, blk, 0, stream>>>(h3, batch, pooled, 2 * HH);

    // ---------------- lin1: [512 x 192] @ [192 x 192] + bias, ReLU ----------------
    dim3 gemmGridL(GG / 16, 3);                // 3 blocks of 64 cols
    gemm_wmma_fused<3 * HH, true><<<gemmGridL, gemmBlk, 0, stream>>>(
        pooled, lin1_w, nullptr, lin1_b, q, 3 * HH);

    // ---------------- lin2 + log_softmax ----------------
    head_kernel<<<dim3((GG + TB - 1) / TB), blk, 0, stream>>>(
        q, lin2_w, lin2_b, (float*)d_out);

    (void)in_sizes; (void)n_in; (void)out_size; (void)ws_size;
}